// AGDN_16638703304814
// MI455X (gfx1250) — compile-verified
//
#include <hip/hip_runtime.h>
#include <math.h>
#include <float.h>

typedef float v2f __attribute__((ext_vector_type(2)));
typedef float v8f __attribute__((ext_vector_type(8)));

#define HC 64           // heads * channels, same both layers
#define SLOPE 0.2f

__device__ __forceinline__ float lrelu(float x) { return x > 0.f ? x : SLOPE * x; }

// float atomic max via monotonic int/uint ordering trick (works for mixed signs)
__device__ __forceinline__ void atomicMaxF(float* addr, float val) {
    if (val >= 0.f) atomicMax((int*)addr, __float_as_int(val));
    else            atomicMin((unsigned int*)addr, __float_as_uint(val));
}

// ---------------------------------------------------------------------------
// Fused GEMM: Y = X @ W and (if DUAL) Yr = X @ Wr.
// X: [n, K] row-major, W/Wr: [K, 64] row-major. One wave per (16*RT)-row tile;
// RT row-tiles share each B load (B traffic dominates A traffic here).
// Uses V_WMMA_F32_16X16X4_F32 (full fp32 — GEMM is <5% of runtime, exact
// numerics beat low-precision throughput).
// A 16x4 f32 layout: lanes 0-15 -> M=0..15 K={0,1}; lanes 16-31 -> K={2,3}.
// B 4x16 f32 layout (mirror): lanes 0-15 -> N, K={0,1}; lanes 16-31 -> K={2,3}.
// C/D 16x16 f32: VGPR r -> M = r + 8*(lane>=16), N = lane&15.
// ---------------------------------------------------------------------------
template <bool DUAL, int RT>
__global__ __launch_bounds__(32)
void gemm_x2(const float* __restrict__ X, const float* __restrict__ W,
             const float* __restrict__ Wr, float* __restrict__ Y,
             float* __restrict__ Yr, int n, int K) {
    const int lane  = threadIdx.x;
    const int row16 = lane & 15;
    const int khalf = (lane >> 4) << 1;   // 0 or 2
    const int col   = lane & 15;
    const int r0    = blockIdx.x * (16 * RT);

    int arow[RT];
    #pragma unroll
    for (int t = 0; t < RT; ++t) {
        int r = r0 + 16 * t + row16;
        arow[t] = r < n ? r : n - 1;      // clamp (discarded at store)
    }

    v8f acc[RT][4]  = {};
    v8f accR[RT][4] = {};

    #pragma unroll 2
    for (int k0 = 0; k0 < K; k0 += 4) {
        v2f a[RT];
        #pragma unroll
        for (int t = 0; t < RT; ++t)
            a[t] = *reinterpret_cast<const v2f*>(X + (size_t)arow[t] * K + k0 + khalf);

        #pragma unroll
        for (int j = 0; j < 4; ++j) {
            v2f b;
            const float* bp = W + (size_t)(k0 + khalf) * HC + j * 16 + col;
            b.x = bp[0]; b.y = bp[HC];
            #pragma unroll
            for (int t = 0; t < RT; ++t)
                acc[t][j] = __builtin_amdgcn_wmma_f32_16x16x4_f32(
                    false, a[t], false, b, (short)0, acc[t][j], false, false);
            if (DUAL) {
                v2f br;
                const float* brp = Wr + (size_t)(k0 + khalf) * HC + j * 16 + col;
                br.x = brp[0]; br.y = brp[HC];
                #pragma unroll
                for (int t = 0; t < RT; ++t)
                    accR[t][j] = __builtin_amdgcn_wmma_f32_16x16x4_f32(
                        false, a[t], false, br, (short)0, accR[t][j], false, false);
            }
        }
    }

    const int mhi = (lane >> 4) * 8;
    #pragma unroll
    for (int t = 0; t < RT; ++t)
        #pragma unroll
        for (int j = 0; j < 4; ++j)
            #pragma unroll
            for (int r = 0; r < 8; ++r) {
                int m = r0 + 16 * t + mhi + r;
                if (m < n) {
                    Y[(size_t)m * HC + j * 16 + col] = acc[t][j][r];
                    if (DUAL) Yr[(size_t)m * HC + j * 16 + col] = accR[t][j][r];
                }
            }
}

// ---------------------------------------------------------------------------
// Per-node prep, layer 1 (H=4, C=16): al/ar coefficients + init max/sum/agg.
// ---------------------------------------------------------------------------
__global__ void node_prep1(const float* __restrict__ xlin,
                           const float* __restrict__ attL, const float* __restrict__ attR,
                           float* __restrict__ al, float* __restrict__ ar,
                           float* __restrict__ nmax, float* __restrict__ nsum,
                           float* __restrict__ agg, int n) {
    int i = blockIdx.x * blockDim.x + threadIdx.x;
    if (i >= n) return;
    const float4* rowp = (const float4*)(xlin + (size_t)i * 64);
    const float4* lp = (const float4*)attL;
    const float4* rp = (const float4*)attR;
    float4* aggp = (float4*)(agg + (size_t)i * 64);
    float aL[4] = {0, 0, 0, 0}, aR[4] = {0, 0, 0, 0};
    #pragma unroll
    for (int q = 0; q < 16; ++q) {
        float4 v = rowp[q];
        float4 l = lp[q];
        float4 r = rp[q];
        int h = q >> 2;                       // 4 cols per q, 16 cols per head
        aL[h] += v.x * l.x + v.y * l.y + v.z * l.z + v.w * l.w;
        aR[h] += v.x * r.x + v.y * r.y + v.z * r.z + v.w * r.w;
        aggp[q] = make_float4(0.f, 0.f, 0.f, 0.f);
    }
    #pragma unroll
    for (int h = 0; h < 4; ++h) {
        al[i * 4 + h] = aL[h];
        ar[i * 4 + h] = aR[h];
        nmax[i * 4 + h] = -FLT_MAX;
        nsum[i * 4 + h] = 0.f;
    }
}

// Per-node prep, layer 2 (H=1, C=64).
__global__ void node_prep2(const float* __restrict__ xlin,
                           const float* __restrict__ attL, const float* __restrict__ attR,
                           float* __restrict__ al, float* __restrict__ ar,
                           float* __restrict__ nmax, float* __restrict__ nsum,
                           float* __restrict__ agg, int n) {
    int i = blockIdx.x * blockDim.x + threadIdx.x;
    if (i >= n) return;
    const float4* rowp = (const float4*)(xlin + (size_t)i * 64);
    const float4* lp = (const float4*)attL;
    const float4* rp = (const float4*)attR;
    float4* aggp = (float4*)(agg + (size_t)i * 64);
    float aL = 0.f, aR = 0.f;
    #pragma unroll
    for (int q = 0; q < 16; ++q) {
        float4 v = rowp[q];
        float4 l = lp[q];
        float4 r = rp[q];
        aL += v.x * l.x + v.y * l.y + v.z * l.z + v.w * l.w;
        aR += v.x * r.x + v.y * r.y + v.z * r.z + v.w * r.w;
        aggp[q] = make_float4(0.f, 0.f, 0.f, 0.f);
    }
    al[i] = aL; ar[i] = aR; nmax[i] = -FLT_MAX; nsum[i] = 0.f;
}

// ---------------------------------------------------------------------------
// Edge passes: segment max, segment exp-sum, then weighted aggregate.
// al/ar/max/sum arrays are tiny (<= 1.6MB) -> L2/WGP$ resident.
// ---------------------------------------------------------------------------
template <int H>
__global__ void edge_max_k(const int* __restrict__ src, const int* __restrict__ dst,
                           const float* __restrict__ al, const float* __restrict__ ar,
                           float* __restrict__ nmax, int e) {
    int i = blockIdx.x * blockDim.x + threadIdx.x;
    if (i >= e) return;
    int s = src[i], d = dst[i];
    #pragma unroll
    for (int h = 0; h < H; ++h) {
        float lg = lrelu(al[s * H + h] + ar[d * H + h]);
        atomicMaxF(&nmax[d * H + h], lg);
    }
}

template <int H>
__global__ void edge_sum_k(const int* __restrict__ src, const int* __restrict__ dst,
                           const float* __restrict__ al, const float* __restrict__ ar,
                           const float* __restrict__ nmax, float* __restrict__ nsum, int e) {
    int i = blockIdx.x * blockDim.x + threadIdx.x;
    if (i >= e) return;
    int s = src[i], d = dst[i];
    #pragma unroll
    for (int h = 0; h < H; ++h) {
        float lg = lrelu(al[s * H + h] + ar[d * H + h]);
        atomicAdd(&nsum[d * H + h], __expf(lg - nmax[d * H + h]));
    }
}

// One 32-lane group per edge (wave32-native); each lane handles 2 of 64 cols.
template <int H>
__global__ void edge_agg_k(const int* __restrict__ src, const int* __restrict__ dst,
                           const float* __restrict__ al, const float* __restrict__ ar,
                           const float* __restrict__ nmax, const float* __restrict__ nsum,
                           const float* __restrict__ xlin, float* __restrict__ agg, int e) {
    int g = blockIdx.x * (blockDim.x >> 5) + (threadIdx.x >> 5);
    int lane = threadIdx.x & 31;
    if (g >= e) return;
    int s = src[g], d = dst[g];
    #pragma unroll
    for (int half = 0; half < 2; ++half) {
        int col = lane + 32 * half;
        int h = (col * H) >> 6;               // col / (64/H)
        float a = __expf(lrelu(al[s * H + h] + ar[d * H + h]) - nmax[d * H + h])
                  / (nsum[d * H + h] + 1e-16f);
        atomicAdd(&agg[(size_t)d * 64 + col], xlin[(size_t)s * 64 + col] * a);
    }
}

// out1 = elu(agg + res + b1), written in place into agg (becomes h).
__global__ void finish1_k(float* __restrict__ agg, const float* __restrict__ res,
                          const float* __restrict__ bias, int total) {
    int i = blockIdx.x * blockDim.x + threadIdx.x;
    if (i >= total) return;
    float v = agg[i] + res[i] + bias[i & 63];
    agg[i] = v > 0.f ? v : (__expf(v) - 1.f);
}

// out = agg + x_lin2 (identity residual) + b2
__global__ void finish2_k(const float* __restrict__ agg, const float* __restrict__ xlin,
                          const float* __restrict__ bias, float* __restrict__ out, int total) {
    int i = blockIdx.x * blockDim.x + threadIdx.x;
    if (i >= total) return;
    out[i] = agg[i] + xlin[i] + bias[i & 63];
}

// ---------------------------------------------------------------------------
extern "C" void kernel_launch(void* const* d_in, const int* in_sizes, int n_in,
                              void* d_out, int out_size, void* d_ws, size_t ws_size,
                              hipStream_t stream) {
    const float* x     = (const float*)d_in[0];
    const int*   eidx  = (const int*)d_in[1];
    const float* W1    = (const float*)d_in[2];
    const float* attL1 = (const float*)d_in[3];
    const float* attR1 = (const float*)d_in[4];
    const float* resW1 = (const float*)d_in[5];
    const float* b1    = (const float*)d_in[6];
    const float* W2    = (const float*)d_in[7];
    const float* attL2 = (const float*)d_in[8];
    const float* attR2 = (const float*)d_in[9];
    const float* b2    = (const float*)d_in[10];

    const int n = in_sizes[0] / 128;
    const int e = in_sizes[1] / 2;
    const int* srcI = eidx;
    const int* dstI = eidx + e;

    // Workspace layout (floats), with cross-layer reuse: ~83 MB total.
    float* ws = (float*)d_ws;
    float* A  = ws;                        // n*64: x_lin1, reused as x_lin2
    float* B  = A  + (size_t)n * 64;       // n*64: x_res1, reused as agg2
    float* C  = B  + (size_t)n * 64;       // n*64: agg1, reused in-place as h
    float* AL = C  + (size_t)n * 64;       // n*4 (layer2 uses first n)
    float* AR = AL + (size_t)n * 4;
    float* MX = AR + (size_t)n * 4;
    float* SM = MX + (size_t)n * 4;

    constexpr int RT = 2;                  // row-tiles per wave
    const int tiles = (n + 16 * RT - 1) / (16 * RT);
    const int nThr  = (n + 255) / 256;
    const int eThr  = (e + 255) / 256;
    const int eGrp  = (e + 7) / 8;         // 8 edge-groups (32 lanes each) / 256-block
    const int fThr  = (n * 64 + 255) / 256;

    // ---- Layer 1 ----
    gemm_x2<true, RT><<<tiles, 32, 0, stream>>>(x, W1, resW1, A, B, n, 128);
    node_prep1<<<nThr, 256, 0, stream>>>(A, attL1, attR1, AL, AR, MX, SM, C, n);
    edge_max_k<4><<<eThr, 256, 0, stream>>>(srcI, dstI, AL, AR, MX, e);
    edge_sum_k<4><<<eThr, 256, 0, stream>>>(srcI, dstI, AL, AR, MX, SM, e);
    edge_agg_k<4><<<eGrp, 256, 0, stream>>>(srcI, dstI, AL, AR, MX, SM, A, C, e);
    finish1_k<<<fThr, 256, 0, stream>>>(C, B, b1, n * 64);   // C := h = elu(...)

    // ---- Layer 2 ----
    gemm_x2<false, RT><<<tiles, 32, 0, stream>>>(C, W2, nullptr, A, nullptr, n, 64);
    node_prep2<<<nThr, 256, 0, stream>>>(A, attL2, attR2, AL, AR, MX, SM, B, n);
    edge_max_k<1><<<eThr, 256, 0, stream>>>(srcI, dstI, AL, AR, MX, e);
    edge_sum_k<1><<<eThr, 256, 0, stream>>>(srcI, dstI, AL, AR, MX, SM, e);
    edge_agg_k<1><<<eGrp, 256, 0, stream>>>(srcI, dstI, AL, AR, MX, SM, A, B, e);
    finish2_k<<<fThr, 256, 0, stream>>>(B, A, b2, (float*)d_out, n * 64);
}